// MTLRLoss_54357106098556
// MI455X (gfx1250) — compile-verified
//
#include <hip/hip_runtime.h>

typedef float v2f __attribute__((ext_vector_type(2)));
typedef float v8f __attribute__((ext_vector_type(8)));

#define NBINS   32
#define THREADS 256
#define BLOCKS  2048
#define FMAX_   3.402823466e38f

// Stage 1: streaming elementwise loss + hierarchical reduction to one partial
// sum per block. The per-element label y is computed branchlessly via an
// augmented split table: g[0]=-FLT_MAX, g[1..31]=splits, g[32..33]=+FLT_MAX,
// so y_j = (g[j+1-e] < t) with no clamping or control flow.
// Wave-level reduction uses V_WMMA_F32_16X16X4_F32 with B = ones, C = 0:
// sum of all 256 D elements == 16 * (sum of all A elements), independent of
// the exact VGPR striping.
__global__ __launch_bounds__(THREADS) void mtlr_partial_kernel(
    const float* __restrict__ outp,    // [N, 32] logits
    const float* __restrict__ times,   // [N]
    const int*   __restrict__ events,  // [N]
    const float* __restrict__ splits,  // [31]
    float*       __restrict__ partials,// [BLOCKS]
    int nq)                            // N * 8 = number of float4 quanta
{
    __shared__ float s_g[34];          // augmented split table
    __shared__ float s_wave[THREADS / 32];

    if (threadIdx.x < 34) {
        float v;
        if (threadIdx.x == 0)          v = -FMAX_;
        else if (threadIdx.x <= NBINS - 1) v = splits[threadIdx.x - 1];
        else                           v = FMAX_;
        s_g[threadIdx.x] = v;
    }
    __syncthreads();

    const float4* outv = reinterpret_cast<const float4*>(outp);
    float acc = 0.0f;
    const int stride = gridDim.x * blockDim.x;

    for (int q = blockIdx.x * blockDim.x + threadIdx.x; q < nq; q += stride) {
        const int   row = q >> 3;          // 8 float4 per 32-bin row
        const int   j0  = (q & 7) << 2;    // first bin of this quad
        const float t   = times[row];
        const int   e   = events[row];
        const int   m0  = j0 + 1 - e;      // augmented index of first element
        const float4 v  = outv[q];

        #pragma unroll
        for (int c = 0; c < 4; ++c) {
            const float x  = (c == 0) ? v.x : (c == 1) ? v.y : (c == 2) ? v.z : v.w;
            const float g  = s_g[m0 + c];          // unconditional LDS load
            const bool  yb = g < t;                // y = 1 iff split < time
            // max(x,0) - x*y == relu(y ? -x : x)   (exact for y in {0,1})
            const float xs = yb ? -x : x;
            const float sp = __logf(1.0f + __expf(-fabsf(x))); // log1p(exp(-|x|))
            acc += fmaxf(xs, 0.0f) + sp;
        }
    }

    // ---- wave32 reduction through the matrix pipe ----
    v2f a; a.x = acc;  a.y = 0.0f;     // A: per-lane partials (rest zeros)
    v2f b; b.x = 1.0f; b.y = 1.0f;     // B: all ones (4x16)
    v8f cz = {};                       // C = 0
    v8f d = __builtin_amdgcn_wmma_f32_16x16x4_f32(
        /*neg_a=*/false, a, /*neg_b=*/false, b,
        /*c_mod=*/(short)0, cz, /*reuse_a=*/false, /*reuse_b=*/false);

    float s = 0.0f;
    #pragma unroll
    for (int i = 0; i < 8; ++i) s += d[i];
    // Sum over all 32 lanes of s == sum of all D elements == 16 * wave_total.
    #pragma unroll
    for (int off = 1; off < 32; off <<= 1) s += __shfl_xor(s, off, 32);
    const float wave_tot = s * 0.0625f;   // exact /16

    const int lid = threadIdx.x & 31;
    const int wid = threadIdx.x >> 5;
    if (lid == 0) s_wave[wid] = wave_tot;
    __syncthreads();
    if (threadIdx.x == 0) {
        float bs = 0.0f;
        #pragma unroll
        for (int i = 0; i < THREADS / 32; ++i) bs += s_wave[i];
        partials[blockIdx.x] = bs;   // deterministic: fixed slot, every call
    }
}

// Stage 2: reduce BLOCKS partials in f64, scale by 1/(N*32).
__global__ __launch_bounds__(256) void mtlr_final_kernel(
    const float* __restrict__ partials, int n,
    float* __restrict__ out, double inv_count)
{
    __shared__ double sh[256];
    double acc = 0.0;
    for (int i = threadIdx.x; i < n; i += 256) acc += (double)partials[i];
    sh[threadIdx.x] = acc;
    __syncthreads();
    #pragma unroll
    for (int s = 128; s > 0; s >>= 1) {
        if (threadIdx.x < s) sh[threadIdx.x] += sh[threadIdx.x + s];
        __syncthreads();
    }
    if (threadIdx.x == 0) out[0] = (float)(sh[0] * inv_count);
}

extern "C" void kernel_launch(void* const* d_in, const int* in_sizes, int n_in,
                              void* d_out, int out_size, void* d_ws, size_t ws_size,
                              hipStream_t stream) {
    const float* outp   = (const float*)d_in[0];  // [N, 32]
    const float* times  = (const float*)d_in[1];  // [N]
    const int*   events = (const int*)d_in[2];    // [N]
    const float* splits = (const float*)d_in[3];  // [31]

    const int nelem = in_sizes[0];      // N * 32
    const int nq    = nelem / 4;        // float4 quanta
    float* partials = (float*)d_ws;     // BLOCKS floats of scratch

    mtlr_partial_kernel<<<BLOCKS, THREADS, 0, stream>>>(
        outp, times, events, splits, partials, nq);

    const double inv = 1.0 / (double)nelem;
    mtlr_final_kernel<<<1, 256, 0, stream>>>(partials, BLOCKS, (float*)d_out, inv);
}